// SimpleModel02_5755256176695
// MI455X (gfx1250) — compile-verified
//
#include <hip/hip_runtime.h>

typedef float v2f __attribute__((ext_vector_type(2)));
typedef float v8f __attribute__((ext_vector_type(8)));

#define IN_DIM  128
#define HID_DIM 64
#define OUT_DIM 16

// LDS group stride (dwords) for swizzled W1: 128 data + 32 pad so the two
// 16-lane halves of a wave read disjoint 32-bank halves (conflict-free b64).
#define W1_GS 160

// ---------------- init: agg = 0, deg = 1 (self-loop) ----------------
__global__ void gcn_init(float* __restrict__ agg, float* __restrict__ deg,
                         int n, int nAgg) {
  int t = blockIdx.x * blockDim.x + threadIdx.x;
  if (t < nAgg) agg[t] = 0.0f;
  if (t < n)   deg[t] = 1.0f;   // self-loop contributes 1 to every dst degree
}

// ---------------- degree accumulation on dst ----------------
__global__ void gcn_deg(const long long* __restrict__ dst,
                        float* __restrict__ deg, long long nE) {
  long long e = (long long)blockIdx.x * blockDim.x + threadIdx.x;
  if (e < nE) unsafeAtomicAdd(&deg[dst[e]], 1.0f);
}

// ---------------- deg^{-1/2} ----------------
__global__ void gcn_dis(const float* __restrict__ deg,
                        float* __restrict__ dis, int n) {
  int t = blockIdx.x * blockDim.x + threadIdx.x;
  if (t < n) dis[t] = rsqrtf(deg[t]);   // deg >= 1 always (self-loops)
}

// ---------------- per-edge norm = dis[src]*dis[dst] ----------------
__global__ void gcn_norm(const long long* __restrict__ src,
                         const long long* __restrict__ dst,
                         const float* __restrict__ dis,
                         float* __restrict__ norm, long long nE) {
  long long e = (long long)blockIdx.x * blockDim.x + threadIdx.x;
  if (e < nE) norm[e] = dis[src[e]] * dis[dst[e]];
}

// ---------------- GEMM1: h = x @ W1 via V_WMMA_F32_16X16X4_F32 ----------------
// One wave per 16-row tile; W1 staged in LDS pre-swizzled so each lane's
// B fragment (K=k, K=k+1 at a fixed column) is one contiguous ds_load_b64.
// Swizzle: sW[(k>>1)*W1_GS + n*2 + (k&1)] = W1[k][n]
__global__ void __launch_bounds__(256) gcn_gemm1(const float* __restrict__ x,
                                                 const float* __restrict__ W1,
                                                 float* __restrict__ h,
                                                 int nTiles) {
  __shared__ float sW[(IN_DIM / 2) * W1_GS];   // 64 groups * 160 = 40KB
  for (int i = threadIdx.x; i < IN_DIM * HID_DIM; i += 256) {
    int k = i >> 6, n = i & 63;
    sW[(k >> 1) * W1_GS + n * 2 + (k & 1)] = W1[i];
  }
  __syncthreads();

  const int wave = threadIdx.x >> 5;
  const int lane = threadIdx.x & 31;
  const int tile = blockIdx.x * 8 + wave;
  if (tile >= nTiles) return;

  const int row = lane & 15;          // A: M index
  const int kh  = (lane >> 4) << 1;   // A/B: K offset 0 or 2 per half-wave
  const int col = lane & 15;          // B/C: N index

  const float* xrow  = x + (size_t)(tile * 16 + row) * IN_DIM;
  const float* wbase = sW + col * 2;
  v8f acc0 = {}, acc1 = {}, acc2 = {}, acc3 = {};

  for (int k0 = 0; k0 < IN_DIM; k0 += 4) {
    const int k = k0 + kh;
    const v2f a = *(const v2f*)(xrow + k);
    const float* wg = wbase + (k >> 1) * W1_GS;
    const v2f b0 = *(const v2f*)(wg + 0);
    const v2f b1 = *(const v2f*)(wg + 32);
    const v2f b2 = *(const v2f*)(wg + 64);
    const v2f b3 = *(const v2f*)(wg + 96);
    acc0 = __builtin_amdgcn_wmma_f32_16x16x4_f32(false, a, false, b0, (short)0, acc0, false, false);
    acc1 = __builtin_amdgcn_wmma_f32_16x16x4_f32(false, a, false, b1, (short)0, acc1, false, false);
    acc2 = __builtin_amdgcn_wmma_f32_16x16x4_f32(false, a, false, b2, (short)0, acc2, false, false);
    acc3 = __builtin_amdgcn_wmma_f32_16x16x4_f32(false, a, false, b3, (short)0, acc3, false, false);
  }

  // C layout: VGPR r holds M = r + 8*(lane>>4), N = lane&15
  const int mofs = (lane >> 4) * 8;
  float* hb = h + (size_t)(tile * 16 + mofs) * HID_DIM + col;
  for (int r = 0; r < 8; ++r) {
    hb[(size_t)r * HID_DIM + 0]  = acc0[r];
    hb[(size_t)r * HID_DIM + 16] = acc1[r];
    hb[(size_t)r * HID_DIM + 32] = acc2[r];
    hb[(size_t)r * HID_DIM + 48] = acc3[r];
  }
}

// ---------------- edge scatter: agg[dst] += h[src] * norm[e] ----------------
// 16 lanes per edge, each handling one float4 chunk of the 64-wide row.
// h and agg are L2-resident (25.6MB each vs 192MB L2): native f32 atomics.
__global__ void gcn_scatter(const long long* __restrict__ src,
                            const long long* __restrict__ dst,
                            const float* __restrict__ norm,
                            const float* __restrict__ h,
                            float* __restrict__ agg, long long nE) {
  long long t = (long long)blockIdx.x * blockDim.x + threadIdx.x;
  long long e = t >> 4;
  if (e >= nE) return;
  const int c4 = (int)(t & 15);
  const long long s = src[e];
  const long long d = dst[e];
  const float nrm = norm[e];
  const float4 v = ((const float4*)(h + s * HID_DIM))[c4];
  float* ap = agg + d * HID_DIM + c4 * 4;
  unsafeAtomicAdd(ap + 0, v.x * nrm);
  unsafeAtomicAdd(ap + 1, v.y * nrm);
  unsafeAtomicAdd(ap + 2, v.z * nrm);
  unsafeAtomicAdd(ap + 3, v.w * nrm);
}

// ---------------- finalize: relu(agg + h*dis^2 + b1) @ W2 + b2 -> log_softmax ----------------
// W2 swizzled in LDS: sW2[(k>>1)*32 + n*2 + (k&1)] = W2[k][n]  (conflict-free b64)
__global__ void __launch_bounds__(256) gcn_finalize(
    const float* __restrict__ h, const float* __restrict__ agg,
    const float* __restrict__ dis, const float* __restrict__ b1,
    const float* __restrict__ W2, const float* __restrict__ b2,
    float* __restrict__ out, int nTiles) {
  __shared__ float sW2[(HID_DIM / 2) * 32];   // 1024 floats
  __shared__ float sB1[HID_DIM];
  for (int i = threadIdx.x; i < HID_DIM * OUT_DIM; i += 256) {
    int k = i >> 4, n = i & 15;
    sW2[(k >> 1) * 32 + n * 2 + (k & 1)] = W2[i];
  }
  if (threadIdx.x < HID_DIM) sB1[threadIdx.x] = b1[threadIdx.x];
  __syncthreads();

  const int wave = threadIdx.x >> 5;
  const int lane = threadIdx.x & 31;
  const int tile = blockIdx.x * 8 + wave;
  if (tile >= nTiles) return;

  const int row = lane & 15;
  const int kh  = (lane >> 4) << 1;
  const int col = lane & 15;
  const int node = tile * 16 + row;

  float dsq = dis[node];
  dsq *= dsq;                       // self-loop norm = deg^{-1}
  const float* hrow = h   + (size_t)node * HID_DIM;
  const float* arow = agg + (size_t)node * HID_DIM;
  const float* wbase = sW2 + col * 2;

  v8f acc = {};
  for (int k0 = 0; k0 < HID_DIM; k0 += 4) {
    const int k = k0 + kh;
    const v2f av = *(const v2f*)(arow + k);
    const v2f hv = *(const v2f*)(hrow + k);
    const v2f bv = *(const v2f*)(sB1 + k);
    v2f a;
    a.x = fmaxf(av.x + hv.x * dsq + bv.x, 0.0f);
    a.y = fmaxf(av.y + hv.y * dsq + bv.y, 0.0f);
    const v2f b = *(const v2f*)(wbase + (k >> 1) * 32);
    acc = __builtin_amdgcn_wmma_f32_16x16x4_f32(false, a, false, b, (short)0, acc, false, false);
  }

  const float bias2 = b2[col];
  const int mofs = (lane >> 4) * 8;
  for (int r = 0; r < 8; ++r) {
    float z = acc[r] + bias2;
    // row m = mofs + r lives across 16 lanes -> butterfly within width 16
    float m = z;
    for (int off = 1; off < 16; off <<= 1)
      m = fmaxf(m, __shfl_xor(m, off, 16));
    float s = __expf(z - m);
    for (int off = 1; off < 16; off <<= 1)
      s += __shfl_xor(s, off, 16);
    const float lse = m + __logf(s);
    out[(size_t)(tile * 16 + mofs + r) * OUT_DIM + col] = z - lse;
  }
}

// ---------------- launcher ----------------
extern "C" void kernel_launch(void* const* d_in, const int* in_sizes, int n_in,
                              void* d_out, int out_size, void* d_ws, size_t ws_size,
                              hipStream_t stream) {
  const float*     x  = (const float*)d_in[0];
  const long long* ei = (const long long*)d_in[1];   // int64 [2, E]
  const float*     W1 = (const float*)d_in[2];
  const float*     b1 = (const float*)d_in[3];
  const float*     W2 = (const float*)d_in[4];
  const float*     b2 = (const float*)d_in[5];
  float* out = (float*)d_out;

  const int       N  = in_sizes[0] / IN_DIM;
  const long long nE = (long long)in_sizes[1] / 2;
  const long long* src = ei;
  const long long* dst = ei + nE;

  char* ws = (char*)d_ws;
  float* h    = (float*)ws;
  float* agg  = (float*)(ws + (size_t)N * HID_DIM * sizeof(float));
  float* deg  = (float*)(ws + (size_t)N * HID_DIM * sizeof(float) * 2);
  float* dis  = deg + N;
  float* norm = dis + N;   // E floats

  const int nAgg = N * HID_DIM;
  gcn_init<<<(nAgg + 255) / 256, 256, 0, stream>>>(agg, deg, N, nAgg);
  gcn_deg<<<(int)((nE + 255) / 256), 256, 0, stream>>>(dst, deg, nE);
  gcn_dis<<<(N + 255) / 256, 256, 0, stream>>>(deg, dis, N);
  gcn_norm<<<(int)((nE + 255) / 256), 256, 0, stream>>>(src, dst, dis, norm, nE);

  const int nTiles = (N + 15) / 16;   // N=100000 -> exactly 6250 tiles
  gcn_gemm1<<<(nTiles + 7) / 8, 256, 0, stream>>>(x, W1, h, nTiles);

  const long long sThreads = nE * 16;
  gcn_scatter<<<(int)((sThreads + 255) / 256), 256, 0, stream>>>(src, dst, norm, h, agg, nE);

  gcn_finalize<<<(nTiles + 7) / 8, 256, 0, stream>>>(h, agg, dis, b1, W2, b2, out, nTiles);
}